// ChannelWiseDivergence_87093346828695
// MI455X (gfx1250) — compile-verified
//
#include <hip/hip_runtime.h>
#include <hip/hip_bf16.h>
#include <math.h>

#define N_T 640
#define G   128
#define HW  (192 * 192)   // 36864
#define BLK 256           // 8 waves of 32

// LDS staging parameters (dice: TDM, kl: async-to-LDS)
#define CH      2048              // floats per chunk per array (8 KB)
#define NCHUNK  (HW / CH)         // 18, exact

typedef __attribute__((ext_vector_type(2))) float        v2f;
typedef __attribute__((ext_vector_type(8))) float        v8f;
typedef __attribute__((ext_vector_type(4))) int          v4i;
typedef __attribute__((ext_vector_type(8))) int          v8i;
typedef __attribute__((ext_vector_type(4))) unsigned int v4u;

#if __has_builtin(__builtin_amdgcn_global_load_async_to_lds_b128)
#define HAVE_ASYNC_LDS 1
#else
#define HAVE_ASYNC_LDS 0
#endif

#if __has_builtin(__builtin_amdgcn_tensor_load_to_lds) && \
    __has_builtin(__builtin_amdgcn_s_wait_tensorcnt)
#define HAVE_TDM 1
#else
#define HAVE_TDM 0
#endif

// ---------------------------------------------------------------------------
// Wave(32)-wide exact fp32 sum using V_WMMA_F32_16X16X4_F32 as the reducer.
// a={v,0}, B=ones => D[m][n] = v_m + v_{m+16}; per-lane sum of the 8 D regs +
// one shfl_xor(16) -> 32-lane total in all lanes. EXEC all ones at call sites.
// ---------------------------------------------------------------------------
__device__ __forceinline__ float wave_sum(float v) {
#if __has_builtin(__builtin_amdgcn_wmma_f32_16x16x4_f32)
  v2f a; a[0] = v;    a[1] = 0.0f;
  v2f b; b[0] = 1.0f; b[1] = 1.0f;
  v8f c = {0.f, 0.f, 0.f, 0.f, 0.f, 0.f, 0.f, 0.f};
  v8f d = __builtin_amdgcn_wmma_f32_16x16x4_f32(
      false, a, false, b, (short)0, c, false, false);
  float s = ((d[0] + d[1]) + (d[2] + d[3])) + ((d[4] + d[5]) + (d[6] + d[7]));
  s += __shfl_xor(s, 16, 32);
  return s;
#else
  #pragma unroll
  for (int off = 16; off > 0; off >>= 1) v += __shfl_xor(v, off, 32);
  return v;
#endif
}

__device__ __forceinline__ float wave_max(float v) {
  #pragma unroll
  for (int off = 16; off > 0; off >>= 1) v = fmaxf(v, __shfl_xor(v, off, 32));
  return v;
}

__device__ __forceinline__ float block_sum(float v, float* lds8) {
  float w = wave_sum(v);
  if ((threadIdx.x & 31) == 0) lds8[threadIdx.x >> 5] = w;
  __syncthreads();
  float total = ((lds8[0] + lds8[1]) + (lds8[2] + lds8[3])) +
                ((lds8[4] + lds8[5]) + (lds8[6] + lds8[7]));
  __syncthreads();
  return total;
}

__device__ __forceinline__ float block_max(float v, float* lds8) {
  float w = wave_max(v);
  if ((threadIdx.x & 31) == 0) lds8[threadIdx.x >> 5] = w;
  __syncthreads();
  float m = lds8[0];
  #pragma unroll
  for (int i = 1; i < 8; ++i) m = fmaxf(m, lds8[i]);
  __syncthreads();
  return m;
}

#if HAVE_ASYNC_LDS
// one b128 async copy: 16 bytes/lane, lands directly in LDS (ASYNCcnt-tracked)
__device__ __forceinline__ void async_cp16(const float* g, float* l) {
  __builtin_amdgcn_global_load_async_to_lds_b128(
      (__attribute__((address_space(1))) v4i*)g,
      (__attribute__((address_space(3))) v4i*)l,
      0, 0);
}

// issue one 2-array chunk: 4 b128 issues per thread, in wave order
__device__ __forceinline__ void issue_chunk(const float* __restrict__ a,
                                            const float* __restrict__ b,
                                            float* __restrict__ lba,
                                            float* __restrict__ lbb,
                                            int c) {
  const int tid = threadIdx.x;
  #pragma unroll
  for (int k = 0; k < CH; k += BLK * 4) {
    const int f = k + tid * 4;
    async_cp16(a + (size_t)c * CH + f, lba + f);
    async_cp16(b + (size_t)c * CH + f, lbb + f);
  }
}
#endif

#if HAVE_TDM
__device__ __forceinline__ unsigned int lds_offset(const void* p) {
  return (unsigned int)(uintptr_t)(__attribute__((address_space(3))) const void*)p;
}

// One TENSOR_LOAD_TO_LDS: DMA a contiguous CH-float (8 KB) chunk into LDS.
// D# per ISA §8: 2D tensor, data_size=4B, tensor_dim0=tile_dim0=CH,
// tensor_dim1=tile_dim1=1, stride0=CH; groups 2/3 zero (dims>2 unused);
// no multicast/pad/iterate/atomic-barrier. Wave-uniform operands -> SGPRs.
__device__ __forceinline__ void tdm_load_chunk(const float* g, void* l) {
  const unsigned long long ga = (unsigned long long)(uintptr_t)g;
  const unsigned int       la = lds_offset(l);
  v4u g0;
  g0[0] = 1u;                                                  // count=1, user mode
  g0[1] = la;                                                  // lds_addr [63:32]
  g0[2] = (unsigned int)ga;                                    // global_addr[31:0]
  g0[3] = (unsigned int)((ga >> 32) & 0x01FFFFFFu) | 0x80000000u; // addr[56:32] | type=2
  v8i g1;
  g1[0] = 0x20000;                 // [17:16] data_size = 2 -> 4 bytes
  g1[1] = (int)(CH << 16);         // tensor_dim0[15:0]  @ [31:16]
  g1[2] = (int)(1u << 16);         // tensor_dim0 hi = 0 | tensor_dim1 = 1 @ [31:16]
  g1[3] = (int)(CH << 16);         // tensor_dim1 hi = 0 | tile_dim0 = CH @ [31:16]
  g1[4] = 1;                       // tile_dim1 = 1, tile_dim2 = 0
  g1[5] = CH;                      // tensor_dim0_stride[31:0]
  g1[6] = 0;
  g1[7] = 0;
  v4i gz = {0, 0, 0, 0};
#if __clang_major__ >= 23
  v8i gz8 = {0, 0, 0, 0, 0, 0, 0, 0};
  __builtin_amdgcn_tensor_load_to_lds(g0, g1, gz, gz, gz8, 0);
#else
  __builtin_amdgcn_tensor_load_to_lds(g0, g1, gz, gz, 0);
#endif
}
#endif

// ---------------------------------------------------------------------------
// Kernel 1: per-proposal dice loss; 189 MB streamed -> the bandwidth limiter.
// gfx1250 path: double-buffered Tensor Data Mover pipeline. Wave 0 issues one
// TENSOR_LOAD_TO_LDS per 8 KB chunk per array (TDM generates all addresses),
// waits tensorcnt<=2 (per-wave in-order completion) while chunk c+1 is in
// flight; the barrier publishes the tile to all waves.
// ---------------------------------------------------------------------------
__global__ void __launch_bounds__(BLK)
dice_kernel(const float* __restrict__ preds_T,
            const float* __restrict__ gt_T,
            float* __restrict__ loss) {
  __shared__ float rI[8], rX[8], rT[8];
  const int n = blockIdx.x;
  const float* __restrict__ x = preds_T + (size_t)n * HW;
  const float* __restrict__ t = gt_T   + (size_t)n * HW;

  float accI = 0.f, accX = 0.f, accT = 0.f;

#if HAVE_TDM || HAVE_ASYNC_LDS
  __shared__ float bufX[2][CH];
  __shared__ float bufT[2][CH];
#endif

#if HAVE_TDM
  const bool leader = (threadIdx.x < 32);  // wave-uniform branch
  if (leader) {
    tdm_load_chunk(x, bufX[0]);
    tdm_load_chunk(t, bufT[0]);
  }
  for (int c = 0; c < NCHUNK; ++c) {
    const int b = c & 1;
    if (leader) {
      if (c + 1 < NCHUNK) {
        tdm_load_chunk(x + (size_t)(c + 1) * CH, bufX[b ^ 1]);
        tdm_load_chunk(t + (size_t)(c + 1) * CH, bufT[b ^ 1]);
        __builtin_amdgcn_s_wait_tensorcnt((short)2);  // chunk c landed
      } else {
        __builtin_amdgcn_s_wait_tensorcnt((short)0);
      }
    }
    __syncthreads();  // publish TDM writes to all waves
    #pragma unroll
    for (int k = 0; k < CH; k += BLK * 4) {
      const int f = k + threadIdx.x * 4;
      const float4 xv = *(const float4*)&bufX[b][f];
      const float4 tv = *(const float4*)&bufT[b][f];
      accI = fmaf(xv.x, tv.x, accI); accX = fmaf(xv.x, xv.x, accX); accT = fmaf(tv.x, tv.x, accT);
      accI = fmaf(xv.y, tv.y, accI); accX = fmaf(xv.y, xv.y, accX); accT = fmaf(tv.y, tv.y, accT);
      accI = fmaf(xv.z, tv.z, accI); accX = fmaf(xv.z, xv.z, accX); accT = fmaf(tv.z, tv.z, accT);
      accI = fmaf(xv.w, tv.w, accI); accX = fmaf(xv.w, xv.w, accX); accT = fmaf(tv.w, tv.w, accT);
    }
    __syncthreads();  // reads done before buffer b is overwritten (chunk c+2)
  }
#elif HAVE_ASYNC_LDS
  issue_chunk(x, t, bufX[0], bufT[0], 0);
  for (int c = 0; c < NCHUNK; ++c) {
    const int b = c & 1;
    if (c + 1 < NCHUNK) {
      issue_chunk(x, t, bufX[b ^ 1], bufT[b ^ 1], c + 1);
      asm volatile("s_wait_asynccnt 0x4" ::: "memory");
    } else {
      asm volatile("s_wait_asynccnt 0x0" ::: "memory");
    }
    __syncthreads();
    #pragma unroll
    for (int k = 0; k < CH; k += BLK * 4) {
      const int f = k + threadIdx.x * 4;
      const float4 xv = *(const float4*)&bufX[b][f];
      const float4 tv = *(const float4*)&bufT[b][f];
      accI = fmaf(xv.x, tv.x, accI); accX = fmaf(xv.x, xv.x, accX); accT = fmaf(tv.x, tv.x, accT);
      accI = fmaf(xv.y, tv.y, accI); accX = fmaf(xv.y, xv.y, accX); accT = fmaf(tv.y, tv.y, accT);
      accI = fmaf(xv.z, tv.z, accI); accX = fmaf(xv.z, xv.z, accX); accT = fmaf(tv.z, tv.z, accT);
      accI = fmaf(xv.w, tv.w, accI); accX = fmaf(xv.w, xv.w, accX); accT = fmaf(tv.w, tv.w, accT);
    }
    __syncthreads();
  }
#else
  for (int i = threadIdx.x; i < HW; i += BLK) {
    float xv = x[i];
    float tv = t[i];
    accI = fmaf(xv, tv, accI);
    accX = fmaf(xv, xv, accX);
    accT = fmaf(tv, tv, accT);
  }
#endif

  float I  = block_sum(accI, rI);
  float Sx = block_sum(accX, rX);
  float St = block_sum(accT, rT);
  if (threadIdx.x == 0)
    loss[n] = 1.0f - 2.0f * I / (Sx + St + 1e-5f);
}

// ---------------------------------------------------------------------------
// Kernel 2: segmented argmin, first-index tie-break (strict '<' scan).
// ---------------------------------------------------------------------------
__global__ void argmin_kernel(const float* __restrict__ loss,
                              const int* __restrict__ gt_inds_T,
                              int* __restrict__ nms) {
  const int g = threadIdx.x;  // 128 threads
  float best = __builtin_inff();
  int bi = 0;
  for (int i = 0; i < N_T; ++i) {
    if (gt_inds_T[i] == g) {
      float l = loss[i];
      if (l < best) { best = l; bi = i; }
    }
  }
  nms[g] = bi;
}

// online logsumexp / KL accumulator update for one (t, s) pair
__device__ __forceinline__ void kl_update(float t, float s, float& mT, float& sT,
                                          float& u, float& mS, float& sS) {
  if (t > mT) { float r = __expf(mT - t); sT *= r; u *= r; mT = t; }
  float e = __expf(t - mT);
  sT += e;
  u = fmaf(e, t - s, u);
  if (s > mS) { sS *= __expf(mS - s); mS = s; }
  sS += __expf(s - mS);
}

// ---------------------------------------------------------------------------
// Kernel 3: channel-wise KL via single-pass online logsumexp, fed by a
// double-buffered GLOBAL_LOAD_ASYNC_TO_LDS_B128 pipeline (ASYNCcnt-tracked).
// KL_g = u/s_T + (lse_S - lse_T); deterministic partials to workspace.
// ---------------------------------------------------------------------------
__global__ void __launch_bounds__(BLK)
kl_kernel(const float* __restrict__ preds_T,
          const float* __restrict__ preds_S,
          const int* __restrict__ nms,
          const int* __restrict__ gt_inds_S,
          float* __restrict__ kl_part) {
  __shared__ float r0[8], r1[8], r2[8], r3[8], r4[8];
  const int g = blockIdx.x;
  const int trow = nms[gt_inds_S[g]];
  const float* __restrict__ pT = preds_T + (size_t)trow * HW;
  const float* __restrict__ pS = preds_S + (size_t)g    * HW;

  float mT = -__builtin_inff(), sT = 0.f, u = 0.f;
  float mS = -__builtin_inff(), sS = 0.f;

#if HAVE_ASYNC_LDS
  __shared__ float bufT2[2][CH];
  __shared__ float bufS2[2][CH];

  issue_chunk(pT, pS, bufT2[0], bufS2[0], 0);
  for (int c = 0; c < NCHUNK; ++c) {
    const int b = c & 1;
    if (c + 1 < NCHUNK) {
      issue_chunk(pT, pS, bufT2[b ^ 1], bufS2[b ^ 1], c + 1);
      asm volatile("s_wait_asynccnt 0x4" ::: "memory");
    } else {
      asm volatile("s_wait_asynccnt 0x0" ::: "memory");
    }
    __syncthreads();
    #pragma unroll 2
    for (int k = 0; k < CH; k += BLK * 4) {
      const int f = k + threadIdx.x * 4;
      const float4 tv = *(const float4*)&bufT2[b][f];
      const float4 sv = *(const float4*)&bufS2[b][f];
      kl_update(tv.x, sv.x, mT, sT, u, mS, sS);
      kl_update(tv.y, sv.y, mT, sT, u, mS, sS);
      kl_update(tv.z, sv.z, mT, sT, u, mS, sS);
      kl_update(tv.w, sv.w, mT, sT, u, mS, sS);
    }
    __syncthreads();
  }
#else
  for (int i = threadIdx.x; i < HW; i += BLK) {
    kl_update(pT[i], pS[i], mT, sT, u, mS, sS);
  }
#endif

  float MT = block_max(mT, r0);
  float MS = block_max(mS, r1);
  float cT = __expf(mT - MT);
  sT *= cT;
  u  *= cT;
  sS *= __expf(mS - MS);

  float SsT = block_sum(sT, r2);
  float Su  = block_sum(u,  r3);
  float SsS = block_sum(sS, r4);

  if (threadIdx.x == 0) {
    float lseT = MT + logf(SsT);
    float lseS = MS + logf(SsS);
    kl_part[g] = Su / SsT + (lseS - lseT);  // TAU = LOSS_WEIGHT = 1
  }
}

// Fixed-order final sum (deterministic; 128 adds).
__global__ void final_sum_kernel(const float* __restrict__ kl_part,
                                 float* __restrict__ out) {
  if (threadIdx.x == 0 && blockIdx.x == 0) {
    float acc = 0.f;
    for (int g = 0; g < G; ++g) acc += kl_part[g];
    out[0] = acc;
  }
}

// ---------------------------------------------------------------------------
// inputs (setup_inputs order): 0 preds_T f32[640*HW], 1 preds_S f32[128*HW],
// 2 im_ind i32[1], 3 gt_T f32[640*HW], 4 gt_S f32[128*HW] (unused by the loss),
// 5 iter i32[1], 6 gt_inds_T i32[640], 7 gt_inds_S i32[128]. out: f32[1].
// ---------------------------------------------------------------------------
extern "C" void kernel_launch(void* const* d_in, const int* in_sizes, int n_in,
                              void* d_out, int out_size, void* d_ws, size_t ws_size,
                              hipStream_t stream) {
  const float* preds_T   = (const float*)d_in[0];
  const float* preds_S   = (const float*)d_in[1];
  const float* gt_T      = (const float*)d_in[3];
  const int*   gt_inds_T = (const int*)d_in[6];
  const int*   gt_inds_S = (const int*)d_in[7];
  float* out = (float*)d_out;

  // workspace layout: loss[640] f32 | nms[128] i32 | kl_part[128] f32
  float* loss    = (float*)d_ws;
  int*   nms     = (int*)((char*)d_ws + N_T * sizeof(float));
  float* kl_part = (float*)((char*)d_ws + N_T * sizeof(float) + G * sizeof(int));

  dice_kernel<<<N_T, BLK, 0, stream>>>(preds_T, gt_T, loss);
  argmin_kernel<<<1, G, 0, stream>>>(loss, gt_inds_T, nms);
  kl_kernel<<<G, BLK, 0, stream>>>(preds_T, preds_S, nms, gt_inds_S, kl_part);
  final_sum_kernel<<<1, 1, 0, stream>>>(kl_part, out);
}